// _ECELoss_72919954752039
// MI455X (gfx1250) — compile-verified
//
#include <hip/hip_runtime.h>
#include <math.h>
#include <stdint.h>

#define ECE_NBINS 15
#define ECE_THREADS 256
#define ECE_WAVES_PER_BLOCK (ECE_THREADS / 32)
#define ECE_COLS 128
#define ECE_SLOTS 4            // async pipeline depth: 3 copies in flight per wave

// ---- CDNA5 async global->LDS path (ASYNCcnt-tracked copy engine) ----------
#if defined(__HIP_DEVICE_COMPILE__) && defined(__gfx1250__)
#if __has_builtin(__builtin_amdgcn_global_load_async_to_lds_b128) && \
    __has_builtin(__builtin_amdgcn_s_wait_asynccnt)
#define ECE_ASYNC 1
#endif
#endif

typedef float v4f __attribute__((ext_vector_type(4)));
typedef int   v4i __attribute__((vector_size(16)));   // matches builtin's V4i param

// float32 bin edges, mimicking jnp.linspace(0,1,16) in fp32
__device__ __forceinline__ float ece_bin_edge(int k) {
  if (k <= 0) return 0.0f;
  if (k >= ECE_NBINS) return 1.0f;
  return (float)k * (1.0f / 15.0f);
}

// bin b such that conf in (edge[b], edge[b+1]]; -1 if none
__device__ __forceinline__ int ece_bin_index(float conf) {
  int b = (int)ceilf(conf * 15.0f) - 1;
  b = b < 0 ? 0 : (b > ECE_NBINS - 1 ? ECE_NBINS - 1 : b);
  while (b > 0 && conf <= ece_bin_edge(b)) --b;
  while (b < ECE_NBINS - 1 && conf > ece_bin_edge(b + 1)) ++b;
  return (conf > ece_bin_edge(b) && conf <= ece_bin_edge(b + 1)) ? b : -1;
}

// One wave32 owns one row of 128 logits; lane holds 4 consecutive columns.
__device__ __forceinline__ void ece_process_row(v4f v, int lane, int lab,
                                                float* s_hist) {
  // max + first-occurrence argmax within the lane's 4 values
  float m = v.x;
  int idx = lane * 4;
  if (v.y > m) { m = v.y; idx = lane * 4 + 1; }
  if (v.z > m) { m = v.z; idx = lane * 4 + 2; }
  if (v.w > m) { m = v.w; idx = lane * 4 + 3; }
  // wave32 butterfly reduction; tie-break on lowest column index (jnp.argmax)
#pragma unroll
  for (int off = 16; off >= 1; off >>= 1) {
    float om = __shfl_xor(m, off, 32);
    int   oi = __shfl_xor(idx, off, 32);
    if (om > m || (om == m && oi < idx)) { m = om; idx = oi; }
  }
  // max-softmax confidence = 1 / sum(exp(x - max))
  float s = __expf(v.x - m) + __expf(v.y - m) + __expf(v.z - m) + __expf(v.w - m);
#pragma unroll
  for (int off = 16; off >= 1; off >>= 1) s += __shfl_xor(s, off, 32);

  if (lane == 0) {
    float conf = 1.0f / s;
    int b = ece_bin_index(conf);
    if (b >= 0) {
      float acc = (idx == lab) ? 1.0f : 0.0f;
      atomicAdd(&s_hist[b], 1.0f);                 // ds_add_f32
      atomicAdd(&s_hist[ECE_NBINS + b], conf);
      atomicAdd(&s_hist[2 * ECE_NBINS + b], acc);
    }
  }
}

__global__ __launch_bounds__(ECE_THREADS)
void ece_rows_kernel(const float* __restrict__ logits,
                     const int* __restrict__ labels,
                     float* __restrict__ bins, int nrows) {
  __shared__ float s_hist[3 * ECE_NBINS];
#ifdef ECE_ASYNC
  __shared__ float s_stage[ECE_WAVES_PER_BLOCK][ECE_SLOTS][ECE_COLS];
#endif
  if (threadIdx.x < 3 * ECE_NBINS) s_hist[threadIdx.x] = 0.0f;
  __syncthreads();

  const int lane = threadIdx.x & 31;
  const int wib  = threadIdx.x >> 5;
  const int gw   = blockIdx.x * ECE_WAVES_PER_BLOCK + wib;
  const int nw   = gridDim.x * ECE_WAVES_PER_BLOCK;

#ifdef ECE_ASYNC
  {
    // 4-slot ring buffer, 3 async copies in flight per wave.  Async loads
    // retire in order per wave, so after issuing row i+3 a wait of
    // ASYNCcnt<=3 guarantees row i's 512B tile is resident in LDS.
    auto issue = [&](int row, int slot) {
      const float* g = logits + (size_t)row * ECE_COLS + lane * 4;
      float* l = &s_stage[wib][slot][lane * 4];
      v4i __attribute__((address_space(1)))* g1 =
          (v4i __attribute__((address_space(1)))*)(uintptr_t)g;
      v4i __attribute__((address_space(3)))* l3 =
          (v4i __attribute__((address_space(3)))*)(uint32_t)(uintptr_t)l;
      __builtin_amdgcn_global_load_async_to_lds_b128(g1, l3, 0, 0);
    };
#pragma unroll
    for (int k = 0; k < ECE_SLOTS - 1; ++k) {
      int r = gw + k * nw;
      if (r < nrows) issue(r, k);
    }
    int i = 0;
    for (int row = gw; row < nrows; row += nw, ++i) {
      int lab = labels[row];                     // uniform dword, overlaps reductions
      int pre = row + (ECE_SLOTS - 1) * nw;
      if (pre < nrows) {
        issue(pre, (i + ECE_SLOTS - 1) & (ECE_SLOTS - 1));
        __builtin_amdgcn_s_wait_asynccnt(ECE_SLOTS - 1);  // oldest slot ready
      } else {
        __builtin_amdgcn_s_wait_asynccnt(0);              // drain in tail
      }
      asm volatile("" ::: "memory");
      v4f v = *(const v4f*)&s_stage[wib][i & (ECE_SLOTS - 1)][lane * 4]; // ds_read_b128
      ece_process_row(v, lane, lab, s_hist);
    }
  }
#else
  // Fallback: direct nontemporal 128-bit loads (streamed once; don't pollute caches)
  for (int row = gw; row < nrows; row += nw) {
    int lab = labels[row];
    const v4f* gp = (const v4f*)(logits + (size_t)row * ECE_COLS) + lane;
    v4f v = __builtin_nontemporal_load(gp);
    ece_process_row(v, lane, lab, s_hist);
  }
#endif

  __syncthreads();
  if (threadIdx.x < 3 * ECE_NBINS)
    atomicAdd(&bins[threadIdx.x], s_hist[threadIdx.x]);
}

__global__ void ece_zero_kernel(float* bins) {
  if (threadIdx.x < 3 * ECE_NBINS) bins[threadIdx.x] = 0.0f;
}

__global__ void ece_final_kernel(const float* __restrict__ bins,
                                 float* __restrict__ out, float inv_n) {
  if (threadIdx.x == 0 && blockIdx.x == 0) {
    float e = 0.0f;
    for (int b = 0; b < ECE_NBINS; ++b) {
      float c  = bins[b];
      float sc = bins[ECE_NBINS + b];
      float sa = bins[2 * ECE_NBINS + b];
      if (c > 0.0f) {
        float safe = fmaxf(c, 1.0f);
        e += fabsf(sc / safe - sa / safe) * (c * inv_n);
      }
    }
    out[0] = e;
  }
}

extern "C" void kernel_launch(void* const* d_in, const int* in_sizes, int n_in,
                              void* d_out, int out_size, void* d_ws, size_t ws_size,
                              hipStream_t stream) {
  const float* logits = (const float*)d_in[0];
  const int*   labels = (const int*)d_in[1];
  float* bins = (float*)d_ws;   // 45 floats: [count[15], sum_conf[15], sum_acc[15]]
  float* out  = (float*)d_out;

  const int nrows = in_sizes[1];            // N = 1,048,576 (logits are N x 128)

  ece_zero_kernel<<<1, 64, 0, stream>>>(bins);

  int blocks = (nrows + ECE_WAVES_PER_BLOCK - 1) / ECE_WAVES_PER_BLOCK;
  if (blocks > 4096) blocks = 4096;         // persistent waves, ~32 rows/wave
  ece_rows_kernel<<<blocks, ECE_THREADS, 0, stream>>>(logits, labels, bins, nrows);

  ece_final_kernel<<<1, 32, 0, stream>>>(bins, out, 1.0f / (float)nrows);
}